// ViFWrapper_2697239462008
// MI455X (gfx1250) — compile-verified
//
#include <hip/hip_runtime.h>
#include <hip/hip_bf16.h>

// ---------------- problem constants (from reference) ----------------
#define Bc   4
#define Tc   2048
#define Dc   2048
#define NVc  576
#define NIc  128
#define Hc   16
#define DHc  128
#define Kc   12
#define DFFc 8192
#define V0c  35
#define I0c  700
#define TAUc 0.8f
#define ALPHA_MID  0.2f
#define ALPHA_DEEP 0.1f

typedef __attribute__((ext_vector_type(2))) float v2f;
typedef __attribute__((ext_vector_type(8))) float v8f;
typedef __attribute__((ext_vector_type(4))) int   v4i;

// ---- gfx1250 async-to-LDS path (guarded; falls back to plain staging) ----
#if defined(__has_builtin)
#if __has_builtin(__builtin_amdgcn_global_load_async_to_lds_b128) && \
    __has_builtin(__builtin_amdgcn_s_wait_asynccnt)
#define HAVE_ASYNC_LDS 1
#endif
#endif
#ifndef HAVE_ASYNC_LDS
#define HAVE_ASYNC_LDS 0
#endif

#if HAVE_ASYNC_LDS
typedef __attribute__((address_space(1))) v4i as1_v4i;
typedef __attribute__((address_space(3))) v4i as3_v4i;
#endif

// ---------------- reductions ----------------
__device__ __forceinline__ float waveSum(float v) {
#pragma unroll
  for (int o = 16; o > 0; o >>= 1) v += __shfl_xor(v, o, 32);
  return v;
}
__device__ __forceinline__ float waveMax(float v) {
#pragma unroll
  for (int o = 16; o > 0; o >>= 1) v = fmaxf(v, __shfl_xor(v, o, 32));
  return v;
}
__device__ __forceinline__ float blockSum(float v, float* red) {
  __syncthreads();
  v = waveSum(v);
  int lane = threadIdx.x & 31, w = threadIdx.x >> 5;
  if (lane == 0) red[w] = v;
  __syncthreads();
  int nw = (blockDim.x + 31) >> 5;
  if (threadIdx.x < 32) {
    float r = (threadIdx.x < nw) ? red[threadIdx.x] : 0.0f;
    r = waveSum(r);
    if (threadIdx.x == 0) red[0] = r;
  }
  __syncthreads();
  return red[0];
}
__device__ __forceinline__ float blockMax(float v, float* red) {
  __syncthreads();
  v = waveMax(v);
  int lane = threadIdx.x & 31, w = threadIdx.x >> 5;
  if (lane == 0) red[w] = v;
  __syncthreads();
  int nw = (blockDim.x + 31) >> 5;
  if (threadIdx.x < 32) {
    float r = (threadIdx.x < nw) ? red[threadIdx.x] : -3.4e38f;
    r = waveMax(r);
    if (threadIdx.x == 0) red[0] = r;
  }
  __syncthreads();
  return red[0];
}

// ---------------- 1) bulk copy hidden -> out ----------------
__global__ void vif_copy_kernel(const float4* __restrict__ src, float4* __restrict__ dst,
                                long long n4) {
  long long i = (long long)blockIdx.x * blockDim.x + threadIdx.x;
  long long stride = (long long)gridDim.x * blockDim.x;
  for (; i < n4; i += stride) dst[i] = src[i];
}

// ---------------- 2) prominence + top-K ----------------
__global__ void vif_prom_topk_kernel(const float* __restrict__ keys,
                                     const int* __restrict__ vision_idx,
                                     int* __restrict__ idxb, int* __restrict__ gposb) {
  int b = blockIdx.x;
  __shared__ float prom[NVc];
  int wave = threadIdx.x >> 5, lane = threadIdx.x & 31, nw = blockDim.x >> 5;
  for (int n = wave; n < NVc; n += nw) {
    float acc = 0.0f;
    for (int h = 0; h < Hc; ++h) {
      const float* p = keys + (((long long)b * Hc + h) * NVc + n) * DHc;
      float s = 0.0f;
      for (int d = lane; d < DHc; d += 32) { float x = p[d]; s += x * x; }
      s = waveSum(s);
      acc += sqrtf(s);
    }
    if (lane == 0) prom[n] = acc * (1.0f / Hc);
  }
  __syncthreads();
  if (threadIdx.x == 0) {
    for (int kk = 0; kk < Kc; ++kk) {
      float best = -3.4e38f; int bi = 0;
      for (int n = 0; n < NVc; ++n)
        if (prom[n] > best) { best = prom[n]; bi = n; }
      prom[bi] = -3.4e38f;
      idxb[b * Kc + kk] = bi;
      gposb[b * Kc + kk] = vision_idx[b * NVc + bi];
    }
  }
}

// ---------------- 3) gather relay rows ----------------
__global__ void vif_gather_kernel(const float* __restrict__ hidden,
                                  const int* __restrict__ gpos, float* __restrict__ R) {
  int r = blockIdx.x, b = blockIdx.y;
  int gp = gpos[b * Kc + r];
  const float* src = hidden + ((long long)b * Tc + gp) * Dc;
  float* dst = R + (long long)(b * Kc + r) * Dc;
  for (int d = threadIdx.x; d < Dc; d += blockDim.x) dst[d] = src[d];
}

// ---------------- LayerNorm (one row per block) ----------------
__global__ void vif_ln_kernel(const float* __restrict__ X, float* __restrict__ Y,
                              const float* __restrict__ g, const float* __restrict__ bt,
                              int Dd) {
  __shared__ float red[32];
  __shared__ float stats[2];
  long long row = blockIdx.x;
  const float* x = X + row * Dd;
  float s = 0.0f;
  for (int d = threadIdx.x; d < Dd; d += blockDim.x) s += x[d];
  s = blockSum(s, red);
  if (threadIdx.x == 0) stats[0] = s / Dd;
  __syncthreads();
  float m = stats[0];
  float v = 0.0f;
  for (int d = threadIdx.x; d < Dd; d += blockDim.x) { float t = x[d] - m; v += t * t; }
  v = blockSum(v, red);
  if (threadIdx.x == 0) stats[1] = rsqrtf(v / Dd + 1e-5f);
  __syncthreads();
  float r = stats[1];
  float* y = Y + row * Dd;
  for (int d = threadIdx.x; d < Dd; d += blockDim.x)
    y[d] = (x[d] - m) * r * g[d] + bt[d];
}

// ---------------- WMMA f32 16x16x4 GEMM, LDS-staged B panel ----------------
// Block = 128 threads (4 waves). Each block: one 16-wide N column, 4 M-tiles
// (one per wave). B panel (KCH x 16) staged cooperatively into LDS with
// gfx1250 async-to-LDS loads (ASYNCcnt) when available.
// fuse: 0 = none, 1 = tanh-GELU on output, 2 = add resid
#define KCH 64

__global__ void vif_gemm_wmma(const float* __restrict__ A, const float* __restrict__ B,
                              float* __restrict__ C, const float* __restrict__ resid,
                              int M, int N, int Kd, int lda, int ldb, int ldc,
                              int rpbA, long long strideA, int fuse) {
  __shared__ float bs[KCH * 16];
  const int tid = threadIdx.x;       // 0..127
  const int lane = tid & 31;
  const int wid = tid >> 5;          // 0..3
  const int tilesM = M >> 4;
  const int mblocks = (tilesM + 3) >> 2;
  const int ncol = blockIdx.x / mblocks;
  const int mb = blockIdx.x % mblocks;
  const int tm = mb * 4 + wid;
  const int col0 = ncol << 4;
  const int lmod = lane & 15, lhalf = lane >> 4;
  const bool active = (tm < tilesM);

  // A fragment base: row tm*16+lmod, k offset 2*lhalf folded in
  const float* arow = A;
  if (active) {
    int ar = tm * 16 + lmod;
    arow = A + (long long)(ar / rpbA) * strideA + (long long)(ar % rpbA) * lda
             + (lhalf << 1);
  }

  // staging: thread (tid) covers B rows tid/4 and tid/4+32, 4 floats each
  const int srow = tid >> 2;           // 0..31
  const int scol = (tid & 3) << 2;     // 0,4,8,12
  const float* gB = B + (long long)srow * ldb + col0 + scol;
  float* lB = &bs[srow * 16 + scol];

  v8f acc = {};
  for (int k0 = 0; k0 < Kd; k0 += KCH) {
    const float* g0 = gB + (long long)k0 * ldb;
#if HAVE_ASYNC_LDS
    __builtin_amdgcn_global_load_async_to_lds_b128(
        (as1_v4i*)g0, (as3_v4i*)lB, 0, 0);
    __builtin_amdgcn_global_load_async_to_lds_b128(
        (as1_v4i*)(g0 + 32ll * ldb), (as3_v4i*)(lB + 32 * 16), 0, 0);
    __builtin_amdgcn_s_wait_asynccnt(0);
#else
    *(float4*)lB = *(const float4*)g0;
    *(float4*)(lB + 32 * 16) = *(const float4*)(g0 + 32ll * ldb);
#endif
    __syncthreads();
    if (active) {
      const float* ap = arow + k0;
#pragma unroll 4
      for (int kk = 0; kk < KCH; kk += 4) {
        v2f a; a.x = ap[kk]; a.y = ap[kk + 1];
        const int br = (kk + (lhalf << 1)) << 4;   // *16 floats per row
        v2f bb; bb.x = bs[br + lmod]; bb.y = bs[br + 16 + lmod];
        acc = __builtin_amdgcn_wmma_f32_16x16x4_f32(false, a, false, bb,
                                                    (short)0, acc, false, false);
      }
    }
    __syncthreads();
  }

  if (active) {
#pragma unroll
    for (int r = 0; r < 8; ++r) {
      int m = tm * 16 + r + (lhalf << 3);
      int n = col0 + lmod;
      float v = acc[r];
      if (fuse == 2) v += resid[(long long)m * ldc + n];
      if (fuse == 1) {
        float x = v;
        float t = tanhf(0.7978845608028654f * (x + 0.044715f * x * x * x));
        v = 0.5f * x * (1.0f + t);
      }
      C[(long long)m * ldc + n] = v;
    }
  }
}

// ---------------- attention: per (b,h), 12 queries x 128 keys ----------------
__global__ void vif_attn_kernel(const float* __restrict__ q, const float* __restrict__ kbuf,
                                const float* __restrict__ vbuf, float* __restrict__ ctx) {
  int b = blockIdx.x, h = blockIdx.y;
  __shared__ float qs[Kc][DHc];
  __shared__ float sc[Kc][NIc];
  int t = threadIdx.x;  // 0..127
  for (int r = 0; r < Kc; ++r)
    qs[r][t] = q[((long long)(b * Kc + r)) * Dc + h * DHc + t];
  __syncthreads();
  const float* krow = kbuf + ((long long)b * NIc + t) * Dc + h * DHc;
  const float scale = 0.08838834764831845f;  // 1/sqrt(128)
  for (int r = 0; r < Kc; ++r) {
    float s = 0.0f;
    for (int d = 0; d < DHc; ++d) s += qs[r][d] * krow[d];
    sc[r][t] = s * scale;
  }
  __syncthreads();
  if (t < Kc) {
    float mx = -3.4e38f;
    for (int k = 0; k < NIc; ++k) mx = fmaxf(mx, sc[t][k]);
    float sum = 0.0f;
    for (int k = 0; k < NIc; ++k) { float e = expf(sc[t][k] - mx); sc[t][k] = e; sum += e; }
    float inv = 1.0f / sum;
    for (int k = 0; k < NIc; ++k) sc[t][k] *= inv;
  }
  __syncthreads();
  for (int r = 0; r < Kc; ++r) {
    float acc = 0.0f;
    for (int k = 0; k < NIc; ++k)
      acc += sc[r][k] * vbuf[((long long)b * NIc + k) * Dc + h * DHc + t];
    ctx[((long long)(b * Kc + r)) * Dc + h * DHc + t] = acc;
  }
}

// ---------------- scatter relay rows into output hidden ----------------
__global__ void vif_scatter_kernel(const float* __restrict__ Rhat,
                                   const int* __restrict__ gpos, float* __restrict__ out) {
  int r = blockIdx.x, b = blockIdx.y;
  int gp = gpos[b * Kc + r];
  const float* src = Rhat + (long long)(b * Kc + r) * Dc;
  float* dst = out + ((long long)b * Tc + gp) * Dc;
  for (int d = threadIdx.x; d < Dc; d += blockDim.x) dst[d] = src[d];
}

// ---------------- softmax + attention reallocation ----------------
__global__ void vif_realloc_kernel(const float* __restrict__ scores, float* __restrict__ out,
                                   int donor_lo, int donor_hi, float alpha) {
  int b = blockIdx.x;
  const float* s = scores + (long long)b * Tc;
  float* o = out + (long long)b * Tc;
  __shared__ float pbuf[Tc];
  __shared__ float red[32];
  const float invTau = 1.0f / TAUc;
  float mx = -3.4e38f;
  for (int t = threadIdx.x; t < Tc; t += blockDim.x) mx = fmaxf(mx, s[t] * invTau);
  mx = blockMax(mx, red);
  float sum = 0.0f;
  for (int t = threadIdx.x; t < Tc; t += blockDim.x) {
    float e = expf(s[t] * invTau - mx);
    pbuf[t] = e; sum += e;
  }
  sum = blockSum(sum, red);
  float inv = 1.0f / sum;
  float rsum = 0.0f, isum = 0.0f;
  for (int t = threadIdx.x; t < Tc; t += blockDim.x) {
    float p = pbuf[t] * inv;
    float removed = (t >= donor_lo && t < donor_hi) ? alpha * p : 0.0f;
    float kept = p - removed;
    pbuf[t] = kept;
    rsum += removed;
    if (t >= I0c && t < I0c + NIc) isum += kept;
  }
  rsum = blockSum(rsum, red);
  isum = blockSum(isum, red);
  float scale = rsum / (isum + 1e-9f);
  for (int t = threadIdx.x; t < Tc; t += blockDim.x) {
    float kept = pbuf[t];
    float add = (t >= I0c && t < I0c + NIc) ? kept * scale : 0.0f;
    o[t] = kept + add;
  }
}

// ---------------- host launcher ----------------
static inline void launch_gemm(const float* A, const float* B, float* C, const float* resid,
                               int M, int N, int Kd, int lda, int ldb, int ldc,
                               int rpbA, long long strideA, int fuse, hipStream_t stream) {
  int tilesM = M / 16;
  int mblocks = (tilesM + 3) / 4;
  int blocks = mblocks * (N / 16);
  vif_gemm_wmma<<<blocks, 128, 0, stream>>>(A, B, C, resid, M, N, Kd, lda, ldb, ldc,
                                            rpbA, strideA, fuse);
}

extern "C" void kernel_launch(void* const* d_in, const int* in_sizes, int n_in,
                              void* d_out, int out_size, void* d_ws, size_t ws_size,
                              hipStream_t stream) {
  const float* hidden      = (const float*)d_in[0];
  const float* mks         = (const float*)d_in[1];
  const float* mid_scores  = (const float*)d_in[2];
  const float* deep_scores = (const float*)d_in[3];
  const float* Wq = (const float*)d_in[4];
  const float* Wk = (const float*)d_in[5];
  const float* Wv = (const float*)d_in[6];
  const float* Wo = (const float*)d_in[7];
  const float* W1 = (const float*)d_in[8];
  const float* W2 = (const float*)d_in[9];
  const float* g1 = (const float*)d_in[10];
  const float* b1 = (const float*)d_in[11];
  const float* g2 = (const float*)d_in[12];
  const float* b2 = (const float*)d_in[13];
  const int* vision_idx = (const int*)d_in[14];

  float* out_hidden = (float*)d_out;
  float* out_mid    = out_hidden + (long long)Bc * Tc * Dc;
  float* out_deep   = out_mid + (long long)Bc * Tc;

  // workspace layout
  char* ws = (char*)d_ws;
  int* idxb  = (int*)ws;              ws += 64 * sizeof(int);
  int* gposb = (int*)ws;              ws += 64 * sizeof(int);
  const long long RROWS = (long long)Bc * Kc;          // 48
  float* R    = (float*)ws; ws += RROWS * Dc * sizeof(float);
  float* hh   = (float*)ws; ws += RROWS * Dc * sizeof(float);
  float* qb   = (float*)ws; ws += RROWS * Dc * sizeof(float);
  float* ctx  = (float*)ws; ws += RROWS * Dc * sizeof(float);
  float* R2   = (float*)ws; ws += RROWS * Dc * sizeof(float);
  float* h2   = (float*)ws; ws += RROWS * Dc * sizeof(float);
  float* Rhat = (float*)ws; ws += RROWS * Dc * sizeof(float);
  float* act  = (float*)ws; ws += RROWS * DFFc * sizeof(float);
  float* kbuf = (float*)ws; ws += (long long)Bc * NIc * Dc * sizeof(float);
  float* vbuf = (float*)ws; ws += (long long)Bc * NIc * Dc * sizeof(float);

  // 1) copy hidden -> out_hidden
  long long n4 = (long long)Bc * Tc * Dc / 4;
  vif_copy_kernel<<<4096, 256, 0, stream>>>((const float4*)hidden, (float4*)out_hidden, n4);

  // 2) prominence + top-K
  vif_prom_topk_kernel<<<Bc, 256, 0, stream>>>(mks, vision_idx, idxb, gposb);

  // 3) gather relay rows, LN1
  vif_gather_kernel<<<dim3(Kc, Bc), 256, 0, stream>>>(hidden, gposb, R);
  vif_ln_kernel<<<(int)RROWS, 256, 0, stream>>>(R, hh, g1, b1, Dc);

  // 4) projections
  // q = LN(R) @ Wq : [48,2048]
  launch_gemm(hh, Wq, qb, nullptr, (int)RROWS, Dc, Dc, Dc, Dc, Dc, (int)RROWS, 0, 0, stream);
  // k = I @ Wk, v = I @ Wv : A rows are [B x NI] slices of hidden at I0
  const float* Ainstr = hidden + (long long)I0c * Dc;
  launch_gemm(Ainstr, Wk, kbuf, nullptr, Bc * NIc, Dc, Dc, Dc, Dc, Dc,
              NIc, (long long)Tc * Dc, 0, stream);
  launch_gemm(Ainstr, Wv, vbuf, nullptr, Bc * NIc, Dc, Dc, Dc, Dc, Dc,
              NIc, (long long)Tc * Dc, 0, stream);

  // 5) attention
  vif_attn_kernel<<<dim3(Bc, Hc), 128, 0, stream>>>(qb, kbuf, vbuf, ctx);

  // 6) R2 = R + ctx @ Wo
  launch_gemm(ctx, Wo, R2, R, (int)RROWS, Dc, Dc, Dc, Dc, Dc, (int)RROWS, 0, 2, stream);

  // 7) MLP: h2 = LN(R2); act = gelu(h2 @ W1); Rhat = R2 + act @ W2
  vif_ln_kernel<<<(int)RROWS, 256, 0, stream>>>(R2, h2, g2, b2, Dc);
  launch_gemm(h2, W1, act, nullptr, (int)RROWS, DFFc, Dc, Dc, DFFc, DFFc,
              (int)RROWS, 0, 1, stream);
  launch_gemm(act, W2, Rhat, R2, (int)RROWS, Dc, DFFc, DFFc, Dc, Dc,
              (int)RROWS, 0, 2, stream);

  // 8) scatter relay rows back into output hidden
  vif_scatter_kernel<<<dim3(Kc, Bc), 256, 0, stream>>>(Rhat, gposb, out_hidden);

  // 9) reallocation passes (masks are fixed contiguous spans)
  vif_realloc_kernel<<<Bc, 256, 0, stream>>>(mid_scores, out_mid,
                                             V0c + NVc / 2, V0c + NVc, ALPHA_MID);
  vif_realloc_kernel<<<Bc, 256, 0, stream>>>(deep_scores, out_deep,
                                             V0c, V0c + NVc, ALPHA_DEEP);
}